// SNPMiSo_68865505624579
// MI455X (gfx1250) — compile-verified
//
#include <hip/hip_runtime.h>
#include <stdint.h>

#define FMP_W        256
#define KA           9
#define NUM_CLASSES  80
#define M_ANCH       (256 * 256 * KA)          /* 589824 */
#define MC           (M_ANCH * NUM_CLASSES)    /* 47185920 */
#define TOPK         1000
#define NBINS        4096
#define CAND_MAX     4096
#define BIN_LO       (-16.0f)
#define BIN_SCALE    (128.0f)
#define CONF_THRESH  0.05f
#define NMS_THRESH   0.6f
#define CTR_CLAMP    32.0f
#define SCALE_CLAMP  4.135166556742356f        /* log(1000/16) */
#define CLASS_OFFSET 1.0e5f
#define TDM_TILE     4096                      /* floats per TDM tile = 16KB */

typedef unsigned u32x4 __attribute__((ext_vector_type(4)));
typedef unsigned u32x8 __attribute__((ext_vector_type(8)));

__constant__ float c_anch_w[KA] = {32,64,32,64,128,64,128,256,128};
__constant__ float c_anch_h[KA] = {32,32,64,64,64,128,128,128,256};

__device__ __forceinline__ int score_bin(float x) {
    float b = (x - BIN_LO) * BIN_SCALE;
    b = fminf(fmaxf(b, 0.0f), (float)(NBINS - 1));
    return (int)b;
}
// order-preserving float->uint (monotone increasing)
__device__ __forceinline__ unsigned order_key(float x) {
    unsigned u = __float_as_uint(x);
    return (u & 0x80000000u) ? ~u : (u | 0x80000000u);
}
__device__ __forceinline__ float order_unkey(unsigned u) {
    unsigned v = (u & 0x80000000u) ? (u & 0x7FFFFFFFu) : ~u;
    return __uint_as_float(v);
}

// Issue one TDM load: TDM_TILE f32 elements from `gsrc` into LDS byte addr `lds_addr`.
// D# packing per CDNA5 ISA ch.8 (group0: 128b, group1: 256b; 2-group form => <=2D).
__device__ __forceinline__ void tdm_load_tile(const float* gsrc, unsigned lds_addr) {
    unsigned long long ga = (unsigned long long)(uintptr_t)gsrc;
    u32x4 g0;
    g0.x = 1u;                                           // count=1, user mode
    g0.y = lds_addr;                                     // lds_addr[31:0]
    g0.z = (unsigned)(ga & 0xFFFFFFFFu);                 // global_addr[31:0]
    g0.w = (unsigned)((ga >> 32) & 0x01FFFFFFu)          // global_addr[56:32]
         | 0x80000000u;                                  // type=2 ("image")
    const unsigned td0 = 0x7FFFFFFFu;                    // tensor_dim0: huge (no OOB clip)
    u32x8 g1;
    g1.s0 = 0x00020000u;                                 // data_size=2 (4 bytes)
    g1.s1 = (td0 & 0xFFFFu) << 16;                       // tensor_dim0[15:0]
    g1.s2 = (td0 >> 16) | (1u << 16);                    // tensor_dim0[31:16], tensor_dim1=1
    g1.s3 = ((unsigned)TDM_TILE) << 16;                  // tile_dim0
    g1.s4 = 0u;                                          // tile_dim1/2 unused
    g1.s5 = (unsigned)TDM_TILE;                          // tensor_dim0_stride
    g1.s6 = 0u;
    g1.s7 = 0u;
    asm volatile("tensor_load_to_lds %0, %1" :: "s"(g0), "s"(g1) : "memory");
}

// ---------------- k0: zero histogram + meta ----------------
__global__ void k_init(int* hist) {
    for (int i = threadIdx.x + blockIdx.x * blockDim.x; i < NBINS + 16;
         i += blockDim.x * gridDim.x)
        hist[i] = 0;
}

// ---------------- k1: histogram of logits, TDM double buffer ----------------
// Wave 0 of each block drives the Tensor Data Mover: one tensor_load_to_lds
// per 16KB tile, double-buffered on TENSORcnt; block barriers publish tiles.
__global__ void k_hist(const float* __restrict__ cls, int* __restrict__ hist) {
    __shared__ int   lhist[NBINS];                       // 16KB
    __shared__ float tbuf[2][TDM_TILE];                  // 32KB
    const int tid = threadIdx.x;                         // 256 threads = 8 waves
    for (int i = tid; i < NBINS; i += blockDim.x) lhist[i] = 0;
    __syncthreads();

    const int nTiles = MC / TDM_TILE;                    // 11520, exact
    int s = blockIdx.x;
    const int sStep = gridDim.x;
    const bool wave0 = (tid < 32);
    unsigned lds0 = (unsigned)(uintptr_t)&tbuf[0][0];
    unsigned lds1 = (unsigned)(uintptr_t)&tbuf[1][0];

    if (wave0 && s < nTiles)
        tdm_load_tile(cls + (size_t)s * TDM_TILE, lds0);

    int buf = 0;
    while (s < nTiles) {
        int sn = s + sStep;
        if (wave0) {
            if (sn < nTiles) {
                tdm_load_tile(cls + (size_t)sn * TDM_TILE, buf ? lds0 : lds1);
                __builtin_amdgcn_s_wait_tensorcnt(1);    // older tile landed
            } else {
                __builtin_amdgcn_s_wait_tensorcnt(0);
            }
        }
        __syncthreads();                                 // publish tbuf[buf]
        const float4* p = (const float4*)&tbuf[buf][0];
        #pragma unroll
        for (int k = 0; k < TDM_TILE / 1024; ++k) {      // 4 x float4 per thread
            float4 v = p[tid + k * 256];
            atomicAdd(&lhist[score_bin(v.x)], 1);
            atomicAdd(&lhist[score_bin(v.y)], 1);
            atomicAdd(&lhist[score_bin(v.z)], 1);
            atomicAdd(&lhist[score_bin(v.w)], 1);
        }
        __syncthreads();                                 // reads done before reuse
        buf ^= 1;
        s = sn;
    }
    for (int i = tid; i < NBINS; i += blockDim.x) {
        int c = lhist[i];
        if (c) atomicAdd(&hist[i], c);
    }
}

// ---------------- k2: find threshold bin ----------------
__global__ void k_thresh(const int* __restrict__ hist, int* __restrict__ meta) {
    if (threadIdx.x == 0 && blockIdx.x == 0) {
        int acc = 0, T = 0;
        for (int b = NBINS - 1; b >= 0; --b) {
            acc += hist[b];
            if (acc >= TOPK) { T = b; break; }
        }
        meta[0] = T;
    }
}

// ---------------- k3: compact candidates, async global->LDS double buffer ----
// Per-lane global_load_async_to_lds_b128 stream (ASYNCcnt path), wave-local
// waits, no block barriers in the steady-state loop.
__global__ void k_compact(const float* __restrict__ cls, int* __restrict__ meta,
                          unsigned long long* __restrict__ cand) {
    __shared__ float4 stage[2][256];
    const int tid = threadIdx.x;
    const int T = meta[0];
    const int nStages = MC / 1024;                       // 46080, exact
    int s = blockIdx.x;
    const int sStep = gridDim.x;

    unsigned lds0 = (unsigned)(uintptr_t)&stage[0][tid];
    unsigned lds1 = (unsigned)(uintptr_t)&stage[1][tid];

    if (s < nStages) {
        const float4* src = (const float4*)cls + (size_t)s * 256 + tid;
        asm volatile("global_load_async_to_lds_b128 %0, %1, off"
                     :: "v"(lds0), "v"((unsigned long long)(uintptr_t)src)
                     : "memory");
    }
    int buf = 0;
    while (s < nStages) {
        int sn = s + sStep;
        if (sn < nStages) {
            const float4* src = (const float4*)cls + (size_t)sn * 256 + tid;
            unsigned ldsn = buf ? lds0 : lds1;
            asm volatile("s_wait_dscnt 0\n\t"
                         "global_load_async_to_lds_b128 %0, %1, off\n\t"
                         "s_wait_asynccnt 1"
                         :: "v"(ldsn), "v"((unsigned long long)(uintptr_t)src)
                         : "memory");
        } else {
            asm volatile("s_wait_asynccnt 0" ::: "memory");
        }
        float4 v = stage[buf][tid];
        float xs[4] = {v.x, v.y, v.z, v.w};
        const size_t base = ((size_t)s * 256 + tid) * 4;
        #pragma unroll
        for (int c = 0; c < 4; ++c) {
            if (score_bin(xs[c]) >= T) {
                unsigned idx = (unsigned)(base + c);
                unsigned long long key =
                    ((unsigned long long)order_key(xs[c]) << 32) |
                    (unsigned long long)(0xFFFFFFFFu - idx);
                unsigned pos = atomicAdd((unsigned*)&meta[1], 1u);
                if (pos < CAND_MAX) cand[pos] = key;
            }
        }
        buf ^= 1;
        s = sn;
    }
}

// ---------------- k4: bitonic sort (4096) + top-1000 select + box decode ----
__global__ void k_sort_decode(const float* __restrict__ reg_pred,
                              const int* __restrict__ meta,
                              const unsigned long long* __restrict__ cand,
                              float* __restrict__ tscore,
                              int* __restrict__ tlabel,
                              int* __restrict__ tvalid,
                              float* __restrict__ tbox) {
    __shared__ unsigned long long skey[CAND_MAX];
    const int tid = threadIdx.x;                 // 1024 threads
    int cnt = meta[1];
    if (cnt > CAND_MAX) cnt = CAND_MAX;
    for (int i = tid; i < CAND_MAX; i += 1024)
        skey[i] = (i < cnt) ? cand[i] : 0ULL;    // 0 sorts below all real keys
    __syncthreads();
    // ascending bitonic sort
    for (int k = 2; k <= CAND_MAX; k <<= 1) {
        for (int j = k >> 1; j > 0; j >>= 1) {
            for (int i = tid; i < CAND_MAX; i += 1024) {
                int ixj = i ^ j;
                if (ixj > i) {
                    unsigned long long a = skey[i], b = skey[ixj];
                    bool up = ((i & k) == 0);
                    if (up ? (a > b) : (a < b)) { skey[i] = b; skey[ixj] = a; }
                }
            }
            __syncthreads();
        }
    }
    if (tid < TOPK) {
        unsigned long long key = skey[CAND_MAX - 1 - tid]; // rank tid largest
        bool slot_ok = (tid < cnt);
        float logit = order_unkey((unsigned)(key >> 32));
        unsigned idx = 0xFFFFFFFFu - (unsigned)(key & 0xFFFFFFFFu);
        int aidx = 0, label = 0;
        if (slot_ok && idx < (unsigned)MC) {
            aidx  = (int)(idx / NUM_CLASSES);
            label = (int)(idx % NUM_CLASSES);
        } else {
            slot_ok = false;
        }
        float score = slot_ok ? 1.0f / (1.0f + expf(-logit)) : 0.0f;
        int valid = slot_ok && (score > CONF_THRESH);

        int cell = aidx / KA, k9 = aidx % KA;
        float cx = ((float)(cell % FMP_W) + 0.5f) * 8.0f;
        float cy = ((float)(cell / FMP_W) + 0.5f) * 8.0f;
        float aw = c_anch_w[k9], ah = c_anch_h[k9];
        float4 r = ((const float4*)reg_pred)[aidx];
        float ox = fminf(fmaxf(r.x * aw, -CTR_CLAMP), CTR_CLAMP);
        float oy = fminf(fmaxf(r.y * ah, -CTR_CLAMP), CTR_CLAMP);
        float px = cx + ox, py = cy + oy;
        float w  = aw * expf(fminf(r.z, SCALE_CLAMP));
        float h  = ah * expf(fminf(r.w, SCALE_CLAMP));
        tbox[tid * 4 + 0] = px - 0.5f * w;
        tbox[tid * 4 + 1] = py - 0.5f * h;
        tbox[tid * 4 + 2] = px + 0.5f * w;
        tbox[tid * 4 + 3] = py + 0.5f * h;
        tscore[tid] = score;
        tlabel[tid] = label;
        tvalid[tid] = valid;
    }
}

// ---------------- k5: class-offset greedy NMS + finalize ----------------
__global__ void k_nms(const float* __restrict__ tscore,
                      const int* __restrict__ tlabel,
                      const int* __restrict__ tvalid,
                      const float* __restrict__ tbox,
                      const int* __restrict__ img_h,
                      const int* __restrict__ img_w,
                      float* __restrict__ out) {
    __shared__ float bx[TOPK * 4];
    __shared__ float area[TOPK];
    __shared__ int   keep[TOPK];
    const int tid = threadIdx.x;                 // 1024 threads
    float rb0 = 0, rb1 = 0, rb2 = 0, rb3 = 0, rsc = 0;
    int rlb = 0;
    if (tid < TOPK) {
        rb0 = tbox[tid * 4 + 0]; rb1 = tbox[tid * 4 + 1];
        rb2 = tbox[tid * 4 + 2]; rb3 = tbox[tid * 4 + 3];
        rsc = tscore[tid]; rlb = tlabel[tid];
        float off = (float)rlb * CLASS_OFFSET;
        bx[tid * 4 + 0] = rb0 + off; bx[tid * 4 + 1] = rb1 + off;
        bx[tid * 4 + 2] = rb2 + off; bx[tid * 4 + 3] = rb3 + off;
        area[tid] = fmaxf(rb2 - rb0, 0.0f) * fmaxf(rb3 - rb1, 0.0f);
        keep[tid] = tvalid[tid];
    }
    __syncthreads();
    for (int i = 0; i < TOPK; ++i) {
        if (keep[i]) {
            float x1 = bx[i * 4 + 0], y1 = bx[i * 4 + 1];
            float x2 = bx[i * 4 + 2], y2 = bx[i * 4 + 3];
            float ai = area[i];
            for (int j = tid; j < TOPK; j += 1024) {
                if (j > i && keep[j]) {
                    float lx = fmaxf(x1, bx[j * 4 + 0]);
                    float ly = fmaxf(y1, bx[j * 4 + 1]);
                    float rx = fminf(x2, bx[j * 4 + 2]);
                    float ry = fminf(y2, bx[j * 4 + 3]);
                    float iw = fmaxf(rx - lx, 0.0f), ih = fmaxf(ry - ly, 0.0f);
                    float inter = iw * ih;
                    float iou = inter / fmaxf(ai + area[j] - inter, 1e-10f);
                    if (iou > NMS_THRESH) keep[j] = 0;
                }
            }
        }
        __syncthreads();
    }
    if (tid < TOPK) {
        float fw = (float)img_w[0], fh = (float)img_h[0];
        int kp = keep[tid];
        float b0 = fminf(fmaxf(rb0 / fw, 0.0f), 1.0f);
        float b1 = fminf(fmaxf(rb1 / fh, 0.0f), 1.0f);
        float b2 = fminf(fmaxf(rb2 / fw, 0.0f), 1.0f);
        float b3 = fminf(fmaxf(rb3 / fh, 0.0f), 1.0f);
        out[tid * 4 + 0] = kp ? b0 : 0.0f;
        out[tid * 4 + 1] = kp ? b1 : 0.0f;
        out[tid * 4 + 2] = kp ? b2 : 0.0f;
        out[tid * 4 + 3] = kp ? b3 : 0.0f;
        out[TOPK * 4 + tid] = kp ? rsc : 0.0f;
        ((int*)out)[TOPK * 5 + tid] = kp ? rlb : -1;
    }
}

extern "C" void kernel_launch(void* const* d_in, const int* in_sizes, int n_in,
                              void* d_out, int out_size, void* d_ws, size_t ws_size,
                              hipStream_t stream) {
    const float* cls = (const float*)d_in[0];
    const float* reg = (const float*)d_in[1];
    const int* img_h = (const int*)d_in[2];
    const int* img_w = (const int*)d_in[3];

    int* hist = (int*)d_ws;
    int* meta = hist + NBINS;                                   // [0]=Tbin [1]=count
    unsigned long long* cand = (unsigned long long*)(meta + 16);
    float* tscore = (float*)(cand + CAND_MAX);
    int*   tlabel = (int*)(tscore + TOPK);
    int*   tvalid = tlabel + TOPK;
    float* tbox   = (float*)(tvalid + TOPK);

    k_init<<<17, 256, 0, stream>>>(hist);
    k_hist<<<2048, 256, 0, stream>>>(cls, hist);
    k_thresh<<<1, 32, 0, stream>>>(hist, meta);
    k_compact<<<2048, 256, 0, stream>>>(cls, meta, cand);
    k_sort_decode<<<1, 1024, 0, stream>>>(reg, meta, cand, tscore, tlabel, tvalid, tbox);
    k_nms<<<1, 1024, 0, stream>>>(tscore, tlabel, tvalid, tbox, img_h, img_w,
                                  (float*)d_out);
}